// CenterLoss_61718680043542
// MI455X (gfx1250) — compile-verified
//
#include <hip/hip_runtime.h>
#include <hip/hip_bf16.h>

// CenterLoss on MI455X (gfx1250, wave32).
//
// Exact algebraic reduction: the label mask keeps one distmat column per row,
// all other entries clamp to the constant 1e-12. So we only compute
//   d_b = ||x_b||^2 + ||c_{l_b}||^2 - 2 x_b . c_{l_b}
// for the 4096 (row, label) pairs, using V_WMMA_F32_16X16X4_F32 to produce
// the diagonals of X.X^T, Cg.Cg^T and X.Cg^T per 16-row tile (full fp32,
// matching the reference's expanded-addmm formulation).

#define BATCH 4096
#define NUM_CLASSES 10000
#define FEAT_DIM 512
#define TILES (BATCH / 16)        // 256 tiles of 16 rows
#define WAVES_PER_BLOCK 4

typedef float v2f __attribute__((ext_vector_type(2)));
typedef float v8f __attribute__((ext_vector_type(8)));

__global__ __launch_bounds__(32 * WAVES_PER_BLOCK)
void center_loss_partial_kernel(const float* __restrict__ x,
                                const float* __restrict__ centers,
                                const int* __restrict__ labels,
                                float* __restrict__ partials)
{
    const int wave = threadIdx.x >> 5;
    const int lane = threadIdx.x & 31;
    const int tile = blockIdx.x * WAVES_PER_BLOCK + wave;
    const int row0 = tile * 16;

    // A-fragment (16x4 f32): lanes 0-15 hold (M=lane, K=k+0..1),
    // lanes 16-31 hold (M=lane-16, K=k+2..3). B-fragment (4x16) of a
    // transposed row-major tile gathers identically per lane.
    const int rowInTile = lane & 15;
    const int koff      = (lane < 16) ? 0 : 2;

    const int label = labels[row0 + rowInTile];
    const float* __restrict__ xp = x       + (size_t)(row0 + rowInTile) * FEAT_DIM + koff;
    const float* __restrict__ cp = centers + (size_t)label              * FEAT_DIM + koff;

    v8f acc_dot = {};   // diag = x_b . c_b
    v8f acc_xx  = {};   // diag = ||x_b||^2
    v8f acc_cc  = {};   // diag = ||c_b||^2

#pragma unroll 4
    for (int k = 0; k < FEAT_DIM; k += 4) {
        v2f a = *(const v2f*)(xp + k);
        v2f b = *(const v2f*)(cp + k);
        // (neg_a, A, neg_b, B, c_mod, C, reuse_a, reuse_b)
        acc_dot = __builtin_amdgcn_wmma_f32_16x16x4_f32(false, a, false, b,
                                                        (short)0, acc_dot, false, false);
        acc_xx  = __builtin_amdgcn_wmma_f32_16x16x4_f32(false, a, false, a,
                                                        (short)0, acc_xx, false, false);
        acc_cc  = __builtin_amdgcn_wmma_f32_16x16x4_f32(false, b, false, b,
                                                        (short)0, acc_cc, false, false);
    }

    // C/D layout: VGPR v, lanes 0-15 -> (M=v, N=lane); lanes 16-31 -> (M=v+8, N=lane-16).
    // Diagonal (m,m): m<8 -> lane m reads acc[m]; m>=8 -> lane 16+m (24..31) reads acc[m-8].
    union { v8f v; float f[8]; } uxx, ucc, udot;
    uxx.v = acc_xx; ucc.v = acc_cc; udot.v = acc_dot;

    float val = 0.0f;
    if (lane < 8) {
        float d = uxx.f[lane] + ucc.f[lane] - 2.0f * udot.f[lane];
        val = fminf(fmaxf(d, 1e-12f), 1e12f);
    } else if (lane >= 24) {
        const int i = lane - 24;
        float d = uxx.f[i] + ucc.f[i] - 2.0f * udot.f[i];
        val = fminf(fmaxf(d, 1e-12f), 1e12f);
    }

    // Deterministic wave32 reduction.
#pragma unroll
    for (int off = 16; off >= 1; off >>= 1)
        val += __shfl_xor(val, off, 32);

    if (lane == 0)
        partials[tile] = val;
}

__global__ __launch_bounds__(TILES)
void center_loss_final_kernel(const float* __restrict__ partials,
                              float* __restrict__ out)
{
    __shared__ float s[TILES];
    const int t = threadIdx.x;
    s[t] = partials[t];
    __syncthreads();
#pragma unroll
    for (int stride = TILES / 2; stride >= 1; stride >>= 1) {
        if (t < stride) s[t] += s[t + stride];
        __syncthreads();
    }
    if (t == 0) {
        // All masked-out entries contribute the clamp floor 1e-12.
        double total = (double)s[0] +
                       (double)BATCH * (double)(NUM_CLASSES - 1) * 1e-12;
        double loss = total / ((double)BATCH * (double)NUM_CLASSES) / (double)BATCH;
        out[0] = (float)loss;
    }
}

extern "C" void kernel_launch(void* const* d_in, const int* in_sizes, int n_in,
                              void* d_out, int out_size, void* d_ws, size_t ws_size,
                              hipStream_t stream) {
    const float* x       = (const float*)d_in[0];   // [4096, 512]
    const float* centers = (const float*)d_in[1];   // [10000, 512]
    const int*   labels  = (const int*)d_in[2];     // [4096]
    float* partials = (float*)d_ws;                 // 256 floats
    float* out      = (float*)d_out;                // 1 float

    center_loss_partial_kernel<<<TILES / WAVES_PER_BLOCK, 32 * WAVES_PER_BLOCK, 0, stream>>>(
        x, centers, labels, partials);
    center_loss_final_kernel<<<1, TILES, 0, stream>>>(partials, out);
}